// StandGAT2_15839839387781
// MI455X (gfx1250) — compile-verified
//
#include <hip/hip_runtime.h>
#include <hip/hip_bf16.h>

typedef __attribute__((ext_vector_type(2))) float v2f;
typedef __attribute__((ext_vector_type(8))) float v8f;

#define NEG_SLOPE 0.2f
#define EPS_GAT 1e-16f

// ---- order-preserving float<->uint encoding for atomicMax on floats ----
__device__ __forceinline__ unsigned encf(float f) {
    unsigned u = __float_as_uint(f);
    return (u & 0x80000000u) ? ~u : (u | 0x80000000u);
}
__device__ __forceinline__ float decf(unsigned u) {
    return (u & 0x80000000u) ? __uint_as_float(u & 0x7FFFFFFFu) : __uint_as_float(~u);
}

__device__ __forceinline__ float lrelu(float x) { return x > 0.f ? x : NEG_SLOPE * x; }

// ------------------------------------------------------------------
// zero fill (grid-stride)
// ------------------------------------------------------------------
__global__ void fill_zero_f32(float* __restrict__ p, long long n) {
    long long i = (long long)blockIdx.x * blockDim.x + threadIdx.x;
    long long s = (long long)gridDim.x * blockDim.x;
    for (; i < n; i += s) p[i] = 0.f;
}

// ------------------------------------------------------------------
// GEMM1: X[M,K] * W[K,Nc] -> H[M,Nc], fp32 WMMA 16x16x4.
// One block = one 16-row tile; 4 waves, each covers 64 columns (4 accs).
// M % 16 == 0 (50000 = 3125*16), K % 4 == 0, Nc % 64 == 0. K,Nc static.
// ------------------------------------------------------------------
template <int K, int Nc>
__global__ __launch_bounds__(128) void gemm1_wmma(
    const float* __restrict__ X, const float* __restrict__ W,
    float* __restrict__ H) {
    const int wave = threadIdx.x >> 5;
    const int lane = threadIdx.x & 31;
    const int half = lane >> 4;
    const int lm   = lane & 15;
    const int mt   = blockIdx.x;
    const int row  = mt * 16 + lm;
    const int col0 = wave * 64 + lm;

    v8f acc0 = {}, acc1 = {}, acc2 = {}, acc3 = {};
    const float* xrow = X + (size_t)row * K;

#pragma unroll 4
    for (int k = 0; k < K; k += 4) {
        const int ka = k + 2 * half;
        v2f a;
        a.x = xrow[ka + 0];
        a.y = xrow[ka + 1];
        const float* w0 = W + (size_t)(ka + 0) * Nc;
        const float* w1 = W + (size_t)(ka + 1) * Nc;
        v2f b0; b0.x = w0[col0 +  0]; b0.y = w1[col0 +  0];
        v2f b1; b1.x = w0[col0 + 16]; b1.y = w1[col0 + 16];
        v2f b2; b2.x = w0[col0 + 32]; b2.y = w1[col0 + 32];
        v2f b3; b3.x = w0[col0 + 48]; b3.y = w1[col0 + 48];
        acc0 = __builtin_amdgcn_wmma_f32_16x16x4_f32(false, a, false, b0, (short)0, acc0, false, false);
        acc1 = __builtin_amdgcn_wmma_f32_16x16x4_f32(false, a, false, b1, (short)0, acc1, false, false);
        acc2 = __builtin_amdgcn_wmma_f32_16x16x4_f32(false, a, false, b2, (short)0, acc2, false, false);
        acc3 = __builtin_amdgcn_wmma_f32_16x16x4_f32(false, a, false, b3, (short)0, acc3, false, false);
    }

#pragma unroll
    for (int r = 0; r < 8; ++r) {
        const int m = mt * 16 + r + 8 * half;
        float* hrow = H + (size_t)m * Nc;
        hrow[col0 +  0] = acc0[r];
        hrow[col0 + 16] = acc1[r];
        hrow[col0 + 32] = acc2[r];
        hrow[col0 + 48] = acc3[r];
    }
}

// ------------------------------------------------------------------
// GEMM2: X[M,K] * W[K,Nc] -> H[M,Nc], Nc=40 (3 n-tiles of 16, guarded)
// block = 96 threads = 3 waves (one per 16-col tile)
// ------------------------------------------------------------------
template <int K, int Nc>
__global__ __launch_bounds__(96) void gemm2_wmma(
    const float* __restrict__ X, const float* __restrict__ W,
    float* __restrict__ H) {
    const int wave = threadIdx.x >> 5;
    const int lane = threadIdx.x & 31;
    const int half = lane >> 4;
    const int lm   = lane & 15;
    const int mt   = blockIdx.x;
    const int row  = mt * 16 + lm;
    const int col  = wave * 16 + lm;
    const bool cok = col < Nc;

    v8f acc = {};
    const float* xrow = X + (size_t)row * K;

#pragma unroll 4
    for (int k = 0; k < K; k += 4) {
        const int ka = k + 2 * half;
        v2f a;
        a.x = xrow[ka + 0];
        a.y = xrow[ka + 1];
        v2f b;
        b.x = cok ? W[(size_t)(ka + 0) * Nc + col] : 0.f;
        b.y = cok ? W[(size_t)(ka + 1) * Nc + col] : 0.f;
        acc = __builtin_amdgcn_wmma_f32_16x16x4_f32(false, a, false, b, (short)0, acc, false, false);
    }

    if (cok) {
#pragma unroll
        for (int r = 0; r < 8; ++r) {
            const int m = mt * 16 + r + 8 * half;
            H[(size_t)m * Nc + col] = acc[r];
        }
    }
}

// ------------------------------------------------------------------
// attention scores, layer1: one wave per (node, head); 64 channels
// ------------------------------------------------------------------
__global__ void scores1_kernel(const float* __restrict__ H1,
                               const float* __restrict__ a_src,
                               const float* __restrict__ a_dst,
                               float* __restrict__ ssrc, float* __restrict__ sdst,
                               int N) {
    const int w    = (int)((blockIdx.x * (long long)blockDim.x + threadIdx.x) >> 5);
    const int lane = threadIdx.x & 31;
    if (w >= N * 4) return;
    const int n = w >> 2, h = w & 3;
    const float* hv = H1 + (size_t)n * 256 + h * 64;
    const float* as = a_src + h * 64;
    const float* ad = a_dst + h * 64;
    float ps = hv[lane] * as[lane] + hv[lane + 32] * as[lane + 32];
    float pd = hv[lane] * ad[lane] + hv[lane + 32] * ad[lane + 32];
#pragma unroll
    for (int off = 16; off > 0; off >>= 1) {
        ps += __shfl_down(ps, off, 32);
        pd += __shfl_down(pd, off, 32);
    }
    if (lane == 0) { ssrc[w] = ps; sdst[w] = pd; }
}

// ------------------------------------------------------------------
// attention scores, layer2: one wave per node; 40 channels, 1 head
// ------------------------------------------------------------------
__global__ void scores2_kernel(const float* __restrict__ H2,
                               const float* __restrict__ a_src,
                               const float* __restrict__ a_dst,
                               float* __restrict__ ssrc, float* __restrict__ sdst,
                               int N) {
    const int w    = (int)((blockIdx.x * (long long)blockDim.x + threadIdx.x) >> 5);
    const int lane = threadIdx.x & 31;
    if (w >= N) return;
    const float* hv = H2 + (size_t)w * 40;
    float x0 = hv[lane];
    float ps = x0 * a_src[lane];
    float pd = x0 * a_dst[lane];
    if (lane < 8) {
        float x1 = hv[32 + lane];
        ps += x1 * a_src[32 + lane];
        pd += x1 * a_dst[32 + lane];
    }
#pragma unroll
    for (int off = 16; off > 0; off >>= 1) {
        ps += __shfl_down(ps, off, 32);
        pd += __shfl_down(pd, off, 32);
    }
    if (lane == 0) { ssrc[w] = ps; sdst[w] = pd; }
}

// ------------------------------------------------------------------
// edge pass A: segment max of leaky_relu(s_src[src]+s_dst[dst]) over dst
// virtual edges [E, E+N) are self loops
// ------------------------------------------------------------------
template <int H>
__global__ void edge_max_kernel(const int* __restrict__ src, const int* __restrict__ dst,
                                const float* __restrict__ ssrc, const float* __restrict__ sdst,
                                unsigned* __restrict__ m_u, int E, int N) {
    const long long t = (long long)blockIdx.x * blockDim.x + threadIdx.x;
    const long long total = (long long)(E + N) * H;
    if (t >= total) return;
    const int e = (int)(t / H);
    const int h = (int)(t - (long long)e * H);
    int s, d;
    if (e < E) { s = src[e]; d = dst[e]; } else { s = d = e - E; }
    const float v = lrelu(ssrc[s * H + h] + sdst[d * H + h]);
    atomicMax(&m_u[d * H + h], encf(v));
}

// ------------------------------------------------------------------
// edge pass B: z[dst] += exp(e - m[dst])
// ------------------------------------------------------------------
template <int H>
__global__ void edge_sum_kernel(const int* __restrict__ src, const int* __restrict__ dst,
                                const float* __restrict__ ssrc, const float* __restrict__ sdst,
                                const unsigned* __restrict__ m_u, float* __restrict__ z,
                                int E, int N) {
    const long long t = (long long)blockIdx.x * blockDim.x + threadIdx.x;
    const long long total = (long long)(E + N) * H;
    if (t >= total) return;
    const int e = (int)(t / H);
    const int h = (int)(t - (long long)e * H);
    int s, d;
    if (e < E) { s = src[e]; d = dst[e]; } else { s = d = e - E; }
    const float v = lrelu(ssrc[s * H + h] + sdst[d * H + h]);
    atomicAdd(&z[d * H + h], __expf(v - decf(m_u[d * H + h])));
}

// ------------------------------------------------------------------
// edge pass C: agg[dst, :] += alpha * Hmat[src, :]  (wave per edge)
// lanes 0..H-1 compute alpha once, broadcast via shfl; float4 gather.
// ------------------------------------------------------------------
template <int H, int HD>
__global__ void edge_aggr_kernel(const int* __restrict__ src, const int* __restrict__ dst,
                                 const float* __restrict__ ssrc, const float* __restrict__ sdst,
                                 const unsigned* __restrict__ m_u, const float* __restrict__ z,
                                 const float* __restrict__ Hmat, float* __restrict__ agg,
                                 int E, int N) {
    const int w    = (int)((blockIdx.x * (long long)blockDim.x + threadIdx.x) >> 5);
    const int lane = threadIdx.x & 31;
    if (w >= E + N) return;
    int s, d;
    if (w < E) { s = src[w]; d = dst[w]; } else { s = d = w - E; }

    float alpha = 0.f;
    if (lane < H) {
        const float v = lrelu(ssrc[s * H + lane] + sdst[d * H + lane]);
        alpha = __expf(v - decf(m_u[d * H + lane])) / (z[d * H + lane] + EPS_GAT);
    }

    constexpr int C  = H * HD;   // total channels (multiple of 4)
    constexpr int NV = C / 4;    // float4 count
    const float4* hrow = (const float4*)(Hmat + (size_t)s * C);
    float* arow = agg + (size_t)d * C;
#pragma unroll
    for (int i = lane; i < NV; i += 32) {
        const float4 hv = hrow[i];
        const int head = (i * 4) / HD;           // all 4 channels share a head (4 | HD)
        const float a = __shfl(alpha, head, 32);
        atomicAdd(&arow[4 * i + 0], a * hv.x);
        atomicAdd(&arow[4 * i + 1], a * hv.y);
        atomicAdd(&arow[4 * i + 2], a * hv.z);
        atomicAdd(&arow[4 * i + 3], a * hv.w);
    }
}

// ------------------------------------------------------------------
// bias + relu: h[n,c] = max(agg[n,c] + b[c], 0)
// ------------------------------------------------------------------
__global__ void bias_relu_kernel(const float* __restrict__ agg, const float* __restrict__ b,
                                 float* __restrict__ h, long long n, int C) {
    long long i = (long long)blockIdx.x * blockDim.x + threadIdx.x;
    long long s = (long long)gridDim.x * blockDim.x;
    for (; i < n; i += s) {
        float v = agg[i] + b[(int)(i % C)];
        h[i] = v > 0.f ? v : 0.f;
    }
}

// ------------------------------------------------------------------
// final: out[n,c] = agg2[n,c] + b2[c]
// ------------------------------------------------------------------
__global__ void bias_out_kernel(const float* __restrict__ agg, const float* __restrict__ b,
                                float* __restrict__ out, long long n, int C) {
    long long i = (long long)blockIdx.x * blockDim.x + threadIdx.x;
    long long s = (long long)gridDim.x * blockDim.x;
    for (; i < n; i += s) out[i] = agg[i] + b[(int)(i % C)];
}

extern "C" void kernel_launch(void* const* d_in, const int* in_sizes, int n_in,
                              void* d_out, int out_size, void* d_ws, size_t ws_size,
                              hipStream_t stream) {
    const float* x     = (const float*)d_in[0];
    const int*   adj   = (const int*)d_in[1];
    const float* W1    = (const float*)d_in[2];
    const float* asrc1 = (const float*)d_in[3];
    const float* adst1 = (const float*)d_in[4];
    const float* b1    = (const float*)d_in[5];
    const float* W2    = (const float*)d_in[6];
    const float* asrc2 = (const float*)d_in[7];
    const float* adst2 = (const float*)d_in[8];
    const float* b2    = (const float*)d_in[9];
    float* out = (float*)d_out;

    const int N = in_sizes[0] / 512;      // 50000
    const int E = in_sizes[1] / 2;        // 800000
    const int C1 = 256, H1N = 4, C2 = 40;
    const int* srcv = adj;
    const int* dstv = adj + E;

    // ---- workspace layout (floats) ----
    float* ws = (float*)d_ws;
    float*    h1    = ws;                                   // N*C1
    float*    agg1  = h1 + (size_t)N * C1;                  // N*C1   (zeroed)
    unsigned* m1    = (unsigned*)(agg1 + (size_t)N * C1);   // N*H1N  (zeroed)
    float*    z1    = (float*)m1 + (size_t)N * H1N;         // N*H1N  (zeroed)
    float*    agg2  = z1 + (size_t)N * H1N;                 // N*C2   (zeroed)
    unsigned* m2    = (unsigned*)(agg2 + (size_t)N * C2);   // N      (zeroed)
    float*    z2    = (float*)m2 + (size_t)N;               // N      (zeroed)
    float*    ssrc1 = z2 + (size_t)N;                       // N*H1N
    float*    sdst1 = ssrc1 + (size_t)N * H1N;              // N*H1N
    float*    h2    = sdst1 + (size_t)N * H1N;              // N*C2  (16B-aligned: N*570*4 % 16 == 0)
    float*    ssrc2 = h2 + (size_t)N * C2;                  // N
    float*    sdst2 = ssrc2 + (size_t)N;                    // N

    const long long zeroN = (long long)N * (C1 + H1N + H1N + C2 + 1 + 1);
    fill_zero_f32<<<2048, 256, 0, stream>>>(agg1, zeroN);

    // ---- layer 1 ----
    gemm1_wmma<512, 256><<<N / 16, 128, 0, stream>>>(x, W1, h1);

    {
        const long long waves = (long long)N * H1N;
        scores1_kernel<<<(int)((waves * 32 + 255) / 256), 256, 0, stream>>>(
            h1, asrc1, adst1, ssrc1, sdst1, N);
    }
    {
        const long long tot = (long long)(E + N) * H1N;
        edge_max_kernel<4><<<(int)((tot + 255) / 256), 256, 0, stream>>>(
            srcv, dstv, ssrc1, sdst1, m1, E, N);
        edge_sum_kernel<4><<<(int)((tot + 255) / 256), 256, 0, stream>>>(
            srcv, dstv, ssrc1, sdst1, m1, z1, E, N);
    }
    {
        const long long waves = (long long)(E + N);
        edge_aggr_kernel<4, 64><<<(int)((waves * 32 + 255) / 256), 256, 0, stream>>>(
            srcv, dstv, ssrc1, sdst1, m1, z1, h1, agg1, E, N);
    }
    bias_relu_kernel<<<2048, 256, 0, stream>>>(agg1, b1, h1, (long long)N * C1, C1);

    // ---- layer 2 ----
    gemm2_wmma<256, 40><<<N / 16, 96, 0, stream>>>(h1, W2, h2);

    scores2_kernel<<<(int)(((long long)N * 32 + 255) / 256), 256, 0, stream>>>(
        h2, asrc2, adst2, ssrc2, sdst2, N);
    {
        const long long tot = (long long)(E + N);
        edge_max_kernel<1><<<(int)((tot + 255) / 256), 256, 0, stream>>>(
            srcv, dstv, ssrc2, sdst2, m2, E, N);
        edge_sum_kernel<1><<<(int)((tot + 255) / 256), 256, 0, stream>>>(
            srcv, dstv, ssrc2, sdst2, m2, z2, E, N);
        edge_aggr_kernel<1, 40><<<(int)((tot * 32 + 255) / 256), 256, 0, stream>>>(
            srcv, dstv, ssrc2, sdst2, m2, z2, h2, agg2, E, N);
    }
    bias_out_kernel<<<2048, 256, 0, stream>>>(agg2, b2, out, (long long)N * C2, C2);
}